// ChebNet_58291296141746
// MI455X (gfx1250) — compile-verified
//
#include <hip/hip_runtime.h>

typedef float v2f __attribute__((ext_vector_type(2)));
typedef float v8f __attribute__((ext_vector_type(8)));

#define DEVFN __device__ __forceinline__

DEVFN void atomAddF(float* p, float v) {
#if defined(__gfx1250__)
    unsafeAtomicAdd(p, v);   // native global_atomic_add_f32
#else
    atomicAdd(p, v);
#endif
}

// ---------------------------------------------------------------------------
// Step 1: degree accumulation (self-loops excluded)
// ---------------------------------------------------------------------------
__global__ void deg_kernel(const int* __restrict__ row, const int* __restrict__ col,
                           float* __restrict__ deg, int E) {
    int e = blockIdx.x * blockDim.x + threadIdx.x;
    if (e >= E) return;
    int r = row[e];
    if (r != col[e]) atomAddF(deg + r, 1.0f);
}

// Step 2: dinv (in place over deg) and diag = 2/lmax[batch] - 1
__global__ void node_norm_kernel(float* __restrict__ deg_dinv, const int* __restrict__ batch,
                                 const float* __restrict__ lmax, float* __restrict__ diag, int N) {
    int n = blockIdx.x * blockDim.x + threadIdx.x;
    if (n >= N) return;
    float d = deg_dinv[n];
    deg_dinv[n] = (d > 0.f) ? rsqrtf(d) : 0.f;
    diag[n] = 2.f / lmax[batch[n]] - 1.f;
}

// Step 3: per-edge weight of L_hat off-diagonal
__global__ void edge_w_kernel(const int* __restrict__ row, const int* __restrict__ col,
                              const float* __restrict__ dinv, const int* __restrict__ batch,
                              const float* __restrict__ lmax, float* __restrict__ we, int E) {
    int e = blockIdx.x * blockDim.x + threadIdx.x;
    if (e >= E) return;
    int r = row[e], c = col[e];
    float w = 0.f;
    if (r != c) w = -dinv[r] * dinv[c] * (2.f / lmax[batch[r]]);
    we[e] = w;
}

// ---------------------------------------------------------------------------
// u = diag[:,None] * v   (initializes scatter target, float4)
// ---------------------------------------------------------------------------
template <int F>
__global__ void init_u_kernel(const float* __restrict__ diag, const float* __restrict__ v,
                              float* __restrict__ u, int N) {
    constexpr int CH = F / 4;
    int tid = blockIdx.x * blockDim.x + threadIdx.x;
    if (tid >= N * CH) return;
    int n = tid / CH, c = tid - n * CH;
    float d = diag[n];
    float4 val = *(const float4*)(v + n * F + 4 * c);
    float4 o = make_float4(d * val.x, d * val.y, d * val.z, d * val.w);
    *(float4*)(u + n * F + 4 * c) = o;
}

// ---------------------------------------------------------------------------
// u[row] += we * v[col]   (edge x float4-chunk parallel, HW f32 atomics)
// ---------------------------------------------------------------------------
template <int F>
__global__ void scatter_kernel(const int* __restrict__ row, const int* __restrict__ col,
                               const float* __restrict__ we, const float* __restrict__ v,
                               float* __restrict__ u, int E) {
    constexpr int CH = F / 4;
    int tid = blockIdx.x * blockDim.x + threadIdx.x;
    if (tid >= E * CH) return;
    int e = tid / CH, c = tid - e * CH;
    float w = we[e];
    if (w == 0.f) return;
    int src = col[e], dst = row[e];
    float4 val = *(const float4*)(v + src * F + 4 * c);
    float* up = u + dst * F + 4 * c;
    atomAddF(up + 0, w * val.x);
    atomAddF(up + 1, w * val.y);
    atomAddF(up + 2, w * val.z);
    atomAddF(up + 3, w * val.w);
}

// ---------------------------------------------------------------------------
// Fused Cheb GEMM:  Hout = relu( [V | U] @ [W0 ; W1] + bias )
// f32 WMMA 16x16x4; block = 256 threads = 8 waves; 32-node x 64-col tile.
// A-frag layout (16x4 f32): lanes 0-15 row M, v0=K0 v1=K1; lanes 16-31 v0=K2 v1=K3.
// B-frag layout (4x16 f32): lanes 0-15 col N, v0=K0 v1=K1; lanes 16-31 v0=K2 v1=K3.
// C/D (16x16 f32): VGPR r -> M=r (lanes 0-15) / M=8+r (lanes 16-31), N=lane%16.
// ---------------------------------------------------------------------------
template <int F>
__global__ void __launch_bounds__(256)
cheb_gemm_kernel(const float* __restrict__ V, const float* __restrict__ U,
                 const float* __restrict__ W, const float* __restrict__ bias,
                 float* __restrict__ Hout, int nn) {
    constexpr int K2 = 2 * F;                  // concatenated depth
    __shared__ float As[32][K2 + 1];           // +1 pad: kill LDS bank conflicts
    __shared__ float Ws[K2 * 64];

    const int tid = threadIdx.x;
    const int base = blockIdx.x * 32;

    // Stage A-tile: rows = nodes, cols = [V features | U features]
    for (int i = tid; i < 32 * K2; i += 256) {
        int r = i / K2, k = i - r * K2;
        int node = base + r;
        float val = 0.f;
        if (node < nn) val = (k < F) ? V[node * F + k] : U[node * F + (k - F)];
        As[r][k] = val;
    }
    // Stage weights: W is [2, F, 64] contiguous == [K2 x 64] row-major
    for (int i = tid; i < K2 * 64; i += 256) Ws[i] = W[i];
    __syncthreads();

    const int lane = tid & 31;
    const int wv = tid >> 5;              // 8 waves
    const int rt = wv >> 2;               // row tile 0..1 (16 nodes each)
    const int ct = wv & 3;                // col tile 0..3 (16 cols each)
    const int half = lane >> 4;
    const int idx = lane & 15;
    const int arow = 16 * rt + idx;
    const int colg = 16 * ct + idx;

    v8f acc = {};
#pragma unroll
    for (int k0 = 0; k0 < K2; k0 += 4) {
        const int ka = k0 + 2 * half;
        v2f a, b;
        a.x = As[arow][ka];
        a.y = As[arow][ka + 1];
        b.x = Ws[ka * 64 + colg];
        b.y = Ws[(ka + 1) * 64 + colg];
        acc = __builtin_amdgcn_wmma_f32_16x16x4_f32(
            false, a, false, b, (short)0, acc, false, false);
    }

    const float bv = bias[colg];
#pragma unroll
    for (int r = 0; r < 8; ++r) {
        int node = base + 16 * rt + 8 * half + r;
        if (node < nn) {
            float o = acc[r] + bv;
            Hout[node * 64 + colg] = (o > 0.f) ? o : 0.f;
        }
    }
}

// ---------------------------------------------------------------------------
// Global add pool: g[batch[n]] += h[n]   (float4 chunks, HW f32 atomics)
// ---------------------------------------------------------------------------
__global__ void pool_kernel(const float* __restrict__ h, const int* __restrict__ batch,
                            float* __restrict__ g, int N) {
    int tid = blockIdx.x * blockDim.x + threadIdx.x;
    if (tid >= N * 16) return;
    int n = tid / 16, c = tid - n * 16;
    int b = batch[n];
    float4 val = *(const float4*)(h + n * 64 + 4 * c);
    float* gp = g + b * 64 + 4 * c;
    atomAddF(gp + 0, val.x);
    atomAddF(gp + 1, val.y);
    atomAddF(gp + 2, val.z);
    atomAddF(gp + 3, val.w);
}

// ---------------------------------------------------------------------------
// Tiny MLP head: out = relu(g @ fc1 + b1) @ fc2 + b2   (B=256 rows)
// ---------------------------------------------------------------------------
__global__ void mlp_kernel(const float* __restrict__ g, const float* __restrict__ w1,
                           const float* __restrict__ b1, const float* __restrict__ w2,
                           const float* __restrict__ b2, float* __restrict__ out, int B) {
    int b = blockIdx.x * blockDim.x + threadIdx.x;
    if (b >= B) return;
    float gr[64];
#pragma unroll
    for (int k = 0; k < 64; ++k) gr[k] = g[b * 64 + k];
    float acc = b2[0];
#pragma unroll 4
    for (int j = 0; j < 32; ++j) {
        float s = b1[j];
#pragma unroll
        for (int k = 0; k < 64; ++k) s += gr[k] * w1[k * 32 + j];
        s = (s > 0.f) ? s : 0.f;
        acc += s * w2[j];
    }
    out[b] = acc;
}

// ---------------------------------------------------------------------------
extern "C" void kernel_launch(void* const* d_in, const int* in_sizes, int n_in,
                              void* d_out, int out_size, void* d_ws, size_t ws_size,
                              hipStream_t stream) {
    const int N = in_sizes[0] / 32;   // 100000
    const int E = in_sizes[1] / 2;    // 1600000
    const int B = in_sizes[3];        // 256

    const float* x      = (const float*)d_in[0];
    const int*   ei     = (const int*)d_in[1];
    const int*   row    = ei;
    const int*   col    = ei + E;
    const int*   batch  = (const int*)d_in[2];
    const float* lmax   = (const float*)d_in[3];
    const float* W1     = (const float*)d_in[4];
    const float* b1     = (const float*)d_in[5];
    const float* W2     = (const float*)d_in[6];
    const float* b2     = (const float*)d_in[7];
    const float* W3     = (const float*)d_in[8];
    const float* b3     = (const float*)d_in[9];
    const float* W4     = (const float*)d_in[10];
    const float* b4     = (const float*)d_in[11];
    const float* fc1w   = (const float*)d_in[12];
    const float* fc1b   = (const float*)d_in[13];
    const float* fc2w   = (const float*)d_in[14];
    const float* fc2b   = (const float*)d_in[15];
    float* out = (float*)d_out;

    // Workspace partition (256B aligned)
    char* base = (char*)d_ws;
    size_t off = 0;
    auto alloc = [&](size_t nbytes) {
        void* p = base + off;
        off += (nbytes + 255) & ~(size_t)255;
        return p;
    };
    float* deg  = (float*)alloc((size_t)N * 4);        // becomes dinv in place
    float* diag = (float*)alloc((size_t)N * 4);
    float* we   = (float*)alloc((size_t)E * 4);
    float* u    = (float*)alloc((size_t)N * 64 * 4);
    float* hA   = (float*)alloc((size_t)N * 64 * 4);
    float* hB   = (float*)alloc((size_t)N * 64 * 4);
    float* g    = (float*)alloc((size_t)B * 64 * 4);

    const int T = 256;
    auto cdiv = [](long long a, long long b) { return (int)((a + b - 1) / b); };

    hipMemsetAsync(deg, 0, (size_t)N * 4, stream);
    hipMemsetAsync(g, 0, (size_t)B * 64 * 4, stream);

    deg_kernel<<<cdiv(E, T), T, 0, stream>>>(row, col, deg, E);
    node_norm_kernel<<<cdiv(N, T), T, 0, stream>>>(deg, batch, lmax, diag, N);
    edge_w_kernel<<<cdiv(E, T), T, 0, stream>>>(row, col, deg, batch, lmax, we, E);

    const int gemmBlocks = cdiv(N, 32);

    // Layer 1 (F=32 -> 64)
    init_u_kernel<32><<<cdiv((long long)N * 8, T), T, 0, stream>>>(diag, x, u, N);
    scatter_kernel<32><<<cdiv((long long)E * 8, T), T, 0, stream>>>(row, col, we, x, u, E);
    cheb_gemm_kernel<32><<<gemmBlocks, T, 0, stream>>>(x, u, W1, b1, hA, N);

    // Layer 2 (64 -> 64)
    init_u_kernel<64><<<cdiv((long long)N * 16, T), T, 0, stream>>>(diag, hA, u, N);
    scatter_kernel<64><<<cdiv((long long)E * 16, T), T, 0, stream>>>(row, col, we, hA, u, E);
    cheb_gemm_kernel<64><<<gemmBlocks, T, 0, stream>>>(hA, u, W2, b2, hB, N);

    // Layer 3
    init_u_kernel<64><<<cdiv((long long)N * 16, T), T, 0, stream>>>(diag, hB, u, N);
    scatter_kernel<64><<<cdiv((long long)E * 16, T), T, 0, stream>>>(row, col, we, hB, u, E);
    cheb_gemm_kernel<64><<<gemmBlocks, T, 0, stream>>>(hB, u, W3, b3, hA, N);

    // Layer 4
    init_u_kernel<64><<<cdiv((long long)N * 16, T), T, 0, stream>>>(diag, hA, u, N);
    scatter_kernel<64><<<cdiv((long long)E * 16, T), T, 0, stream>>>(row, col, we, hA, u, E);
    cheb_gemm_kernel<64><<<gemmBlocks, T, 0, stream>>>(hA, u, W4, b4, hB, N);

    // Pool + head
    pool_kernel<<<cdiv((long long)N * 16, T), T, 0, stream>>>(hB, batch, g, N);
    mlp_kernel<<<cdiv(B, T), T, 0, stream>>>(g, fc1w, fc1b, fc2w, fc2b, out, B);
}